// model2_28836410425492
// MI455X (gfx1250) — compile-verified
//
#include <hip/hip_runtime.h>

typedef __attribute__((ext_vector_type(2))) float v2f;
typedef __attribute__((ext_vector_type(8))) float v8f;

#define NB    16
#define NPN   16384
#define NN    (NB*NPN)          // 262144 nodes
#define EE    (NN*8)            // 2097152 edges
#define GXC   22
#define GG    (GXC*GXC)         // 484
#define SS    (NB*GG)           // 7744 cells
#define HSIZE (1u<<22)
#define EPSB  1e-5f

__device__ __forceinline__ float clamp01(float v){ return fminf(fmaxf(v,0.f),1.f); }
__device__ __forceinline__ float eluf(float v){ return v > 0.f ? v : (expf(v)-1.f); }
__device__ __forceinline__ unsigned ordEnc(float f){
  unsigned u = __float_as_uint(f);
  return (u & 0x80000000u) ? ~u : (u | 0x80000000u);
}
__device__ __forceinline__ float ordDec(unsigned o){
  unsigned u = (o & 0x80000000u) ? (o ^ 0x80000000u) : ~o;
  return __uint_as_float(u);
}

// ---------------- precompute ----------------

__global__ void k_basis_deg(const float* __restrict__ ea, const int* __restrict__ dst,
                            float* __restrict__ bas, float* __restrict__ deg, int E_) {
  int e = blockIdx.x*blockDim.x + threadIdx.x;
  if (e >= E_) return;
  float f0 = clamp01(ea[2*e+0]);
  float f1 = clamp01(ea[2*e+1]);
  bas[4*e+0] = (1.f-f0)*(1.f-f1);
  bas[4*e+1] = f0*(1.f-f1);
  bas[4*e+2] = (1.f-f0)*f1;
  bas[4*e+3] = f0*f1;
  unsafeAtomicAdd(&deg[dst[e]], 1.0f);
}

__global__ void k_cells(const float* __restrict__ pos, const int* __restrict__ batch,
                        int* __restrict__ cid, float* __restrict__ cnt,
                        float* __restrict__ ppx, float* __restrict__ ppy, int N_) {
  int n = blockIdx.x*blockDim.x + threadIdx.x;
  if (n >= N_) return;
  const float SXc = 16.0f/346.0f, SYc = 12.0f/260.0f;
  float px = pos[2*n], py = pos[2*n+1];
  int cx = min(max((int)floorf(px/SXc),0),GXC-1);
  int cy = min(max((int)floorf(py/SYc),0),GXC-1);
  int c = batch[n]*GG + cy*GXC + cx;
  cid[n] = c;
  unsafeAtomicAdd(&cnt[c], 1.0f);
  unsafeAtomicAdd(&ppx[c], px);
  unsafeAtomicAdd(&ppy[c], py);
}

__global__ void k_pp(const float* __restrict__ cnt, float* __restrict__ ppx,
                     float* __restrict__ ppy, float* __restrict__ wv,
                     float* __restrict__ tw, int S_) {
  int c = blockIdx.x*blockDim.x + threadIdx.x;
  if (c >= S_) return;
  float ct = cnt[c];
  float d = fmaxf(ct, 1.0f);
  ppx[c] /= d;  ppy[c] /= d;
  float w = (ct > 0.f) ? 1.f : 0.f;
  wv[c] = w;
  unsafeAtomicAdd(tw, w);
}

__global__ void k_edgemask(const int* __restrict__ src, const int* __restrict__ dst,
                           const int* __restrict__ cid, int* __restrict__ csrc,
                           int* __restrict__ cdst, float* __restrict__ emask,
                           unsigned* __restrict__ hash, const float* __restrict__ ppx,
                           const float* __restrict__ ppy, unsigned* __restrict__ mxb, int E_) {
  int e = blockIdx.x*blockDim.x + threadIdx.x;
  if (e >= E_) return;
  int cs = cid[src[e]], cd = cid[dst[e]];
  csrc[e] = cs;  cdst[e] = cd;
  float em = 0.f;
  if (cs != cd) {
    unsigned key = (unsigned)(cs*SS + cd);
    unsigned h = (key * 2654435761u) & (HSIZE-1u);
    for (;;) {
      unsigned old = atomicCAS(&hash[h], 0xFFFFFFFFu, key);
      if (old == 0xFFFFFFFFu) { em = 1.f; break; }
      if (old == key) { break; }
      h = (h + 1u) & (HSIZE-1u);
    }
  }
  emask[e] = em;
  if (em != 0.f) {
    float ax = fabsf(ppx[cs]-ppx[cd]);
    float ay = fabsf(ppy[cs]-ppy[cd]);
    atomicMax(mxb, __float_as_uint(fmaxf(ax, ay)));  // non-negative: bit order == float order
  }
}

__global__ void k_bas2(const int* __restrict__ csrc, const int* __restrict__ cdst,
                       const float* __restrict__ emask, const float* __restrict__ ppx,
                       const float* __restrict__ ppy, const unsigned* __restrict__ mxb,
                       float* __restrict__ bas2, float* __restrict__ deg2, int E_) {
  int e = blockIdx.x*blockDim.x + threadIdx.x;
  if (e >= E_) return;
  float mx = __uint_as_float(*mxb);
  float inv = 1.f / (2.f*mx + 1e-12f);
  int cs = csrc[e], cd = cdst[e];
  float f0 = clamp01((ppx[cs]-ppx[cd])*inv + 0.5f);
  float f1 = clamp01((ppy[cs]-ppy[cd])*inv + 0.5f);
  bas2[4*e+0] = (1.f-f0)*(1.f-f1);
  bas2[4*e+1] = f0*(1.f-f1);
  bas2[4*e+2] = (1.f-f0)*f1;
  bas2[4*e+3] = f0*f1;
  float em = emask[e];
  if (em != 0.f) unsafeAtomicAdd(&deg2[cd], em);
}

// -------- node transform: Y[rows, 4*COUT] = X @ concat_k(W_k), fp32 WMMA, W staged in LDS --------

template<int CIN, int COUT>
__global__ void __launch_bounds__(128)
k_transform_t(const float* __restrict__ X, const float* __restrict__ W,
              float* __restrict__ Y, int rows) {
  constexpr int KSTEPS = (CIN + 3) / 4;
  constexpr int KPAD   = KSTEPS * 4;            // zero-padded K extent
  constexpr int LD     = 4 * COUT;
  __shared__ float wlds[4 * KPAD * COUT];

  for (int i = threadIdx.x; i < 4*KPAD*COUT; i += 128) {
    int kb = i / (KPAD*COUT);
    int r  = i - kb*(KPAD*COUT);
    int k  = r / COUT;
    int j  = r - k*COUT;
    wlds[i] = (k < CIN) ? W[kb*CIN*COUT + k*COUT + j] : 0.f;
  }
  __syncthreads();

  int wave = threadIdx.x >> 5;
  int lane = threadIdx.x & 31;
  int mtile = blockIdx.x*4 + wave;
  if (mtile*16 >= rows) return;
  int row0 = mtile*16;
  int hi  = lane >> 4;                 // 0: K pair {k0,k0+1}; 1: {k0+2,k0+3}
  int l15 = lane & 15;
  int ncol = blockIdx.y*16 + l15;      // B/D column
  int kb = ncol / COUT;                // power-of-2 -> shift
  int jc = ncol - kb*COUT;
  const float* Wc = wlds + kb*(KPAD*COUT) + jc;
  const float* Xr = X + (size_t)(row0 + l15)*CIN;

  v8f acc = {};
#pragma unroll
  for (int ks = 0; ks < KSTEPS; ++ks) {
    int k0 = ks*4 + hi*2;
    v2f a, b;
    // branchless in-range load + select (folds to plain loads when CIN%4==0)
    float ax = Xr[min(k0,   CIN-1)];
    float ay = Xr[min(k0+1, CIN-1)];
    a.x = (k0   < CIN) ? ax : 0.f;
    a.y = (k0+1 < CIN) ? ay : 0.f;
    b.x = Wc[(size_t)k0*COUT];
    b.y = Wc[(size_t)(k0+1)*COUT];
    acc = __builtin_amdgcn_wmma_f32_16x16x4_f32(false, a, false, b, (short)0, acc, false, false);
  }
  int mbase = row0 + hi*8;
#pragma unroll
  for (int r = 0; r < 8; ++r)
    Y[(size_t)(mbase + r)*LD + ncol] = acc[r];
}

// -------- edge aggregate: 32/COUT edges per wave, lane per channel, float4 basis --------

template<int COUT, bool MASKED>
__global__ void __launch_bounds__(256)
k_edgeagg_t(const int* __restrict__ src, const int* __restrict__ dst,
            const float4* __restrict__ bas4, const float* __restrict__ Y,
            float* __restrict__ agg, const float* __restrict__ emask, int E_) {
  constexpr int LD  = 4*COUT;
  constexpr int EPW = 32/COUT;                  // edges per wave
  int lane = threadIdx.x & 31;
  int sub  = lane / COUT;
  int c    = lane - sub*COUT;
  int wid  = (blockIdx.x*blockDim.x + threadIdx.x) >> 5;
  int nw   = (gridDim.x*blockDim.x) >> 5;
  int ngrp = (E_ + EPW - 1)/EPW;
  for (int g = wid; g < ngrp; g += nw) {
    int e = g*EPW + sub;
    if (EPW > 1 && e >= E_) continue;
    if (MASKED) { if (emask[e] == 0.f) continue; }
    int s = src[e], d = dst[e];
    const float* ys = Y + (size_t)s*LD;
    float4 b = bas4[e];
    float m = b.x*ys[c] + b.y*ys[COUT+c] + b.z*ys[2*COUT+c] + b.w*ys[3*COUT+c];
    unsafeAtomicAdd(&agg[(size_t)d*COUT + c], m);
  }
}

// -------- post: deg-normalize + ELU + stat reduce (LDS ds_add_f32) --------

__global__ void __launch_bounds__(256)
k_post1(const float* __restrict__ agg, const float* __restrict__ deg,
        float* __restrict__ hout, float* __restrict__ stats,
        const float* __restrict__ wv, int rows, int cout, int lc) {
  __shared__ float ls[32], lq[32];
  if (threadIdx.x < 32) { ls[threadIdx.x] = 0.f; lq[threadIdx.x] = 0.f; }
  __syncthreads();
  int idx = blockIdx.x*blockDim.x + threadIdx.x;
  if (idx < (rows << lc)) {
    int nrow = idx >> lc;
    int c = idx & (cout-1);
    float d = fmaxf(deg[nrow], 1.0f);
    float a = eluf(agg[idx] / d);
    hout[idx] = a;
    float w = wv ? wv[nrow] : 1.f;
    atomicAdd(&ls[c], w*a);
    atomicAdd(&lq[c], w*a*a);
  }
  __syncthreads();
  if (threadIdx.x < (unsigned)cout) {
    unsafeAtomicAdd(&stats[threadIdx.x],    ls[threadIdx.x]);
    unsafeAtomicAdd(&stats[64+threadIdx.x], lq[threadIdx.x]);
  }
}

__global__ void __launch_bounds__(256)
k_post2(float* __restrict__ h, const float* __restrict__ stats,
        const float* __restrict__ gvec, const float* __restrict__ bvec,
        const float* __restrict__ sc, const float* __restrict__ twp,
        float denomc, int rows, int cout, int lc) {
  int idx = blockIdx.x*blockDim.x + threadIdx.x;
  if (idx >= (rows << lc)) return;
  int c = idx & (cout-1);
  float denom = twp ? fmaxf(*twp, 1.f) : denomc;
  float m   = stats[c] / denom;
  float var = stats[64+c] / denom - m*m;
  float r = (h[idx] - m) * rsqrtf(var + EPSB) * gvec[c] + bvec[c];
  if (sc) r += sc[idx];
  h[idx] = r;
}

// ---------------- pooling (cout fixed at 32) ----------------

__global__ void k_poolmax(const float* __restrict__ h, const int* __restrict__ cid,
                          unsigned* __restrict__ xpord, int N_) {
  int idx = blockIdx.x*blockDim.x + threadIdx.x;
  if (idx >= (N_<<5)) return;
  int nrow = idx >> 5;
  int c = idx & 31;
  atomicMax(&xpord[(cid[nrow]<<5) + c], ordEnc(h[idx]));
}

__global__ void k_pooldecode(const unsigned* __restrict__ xpord, const float* __restrict__ cnt,
                             float* __restrict__ xp, int S_) {
  int idx = blockIdx.x*blockDim.x + threadIdx.x;
  if (idx >= (S_<<5)) return;
  xp[idx] = (cnt[idx>>5] > 0.f) ? ordDec(xpord[idx]) : 0.f;
}

// ---------------- global mean pool + FC (single-wave WMMA) ----------------

__global__ void k_gm(const float* __restrict__ hp, const float* __restrict__ wv,
                     float* __restrict__ gmn, float* __restrict__ gmd, int S_) {
  int idx = blockIdx.x*blockDim.x + threadIdx.x;
  if (idx >= (S_<<5)) return;
  int cell = idx >> 5;
  int c = idx & 31;
  float w = wv[cell];
  if (w != 0.f) {
    unsafeAtomicAdd(&gmn[((cell/GG)<<5) + c], hp[idx]*w);
    if (c == 0) unsafeAtomicAdd(&gmd[cell/GG], w);
  }
}

__global__ void __launch_bounds__(32)
k_fc(const float* __restrict__ gmn, const float* __restrict__ gmd,
     const float* __restrict__ fcW, float* __restrict__ out) {
  int lane = threadIdx.x;
  int hi = lane >> 4, l15 = lane & 15;
  float invd = 1.f / fmaxf(gmd[l15], 1e-12f);
  v8f acc = {};
#pragma unroll
  for (int ks = 0; ks < 8; ++ks) {
    int k0 = ks*4 + hi*2;
    v2f a, b;
    a.x = gmn[(l15<<5) + k0]   * invd;
    a.y = gmn[(l15<<5) + k0+1] * invd;
    b.x = (l15 < 10) ? fcW[k0*10 + l15]     : 0.f;
    b.y = (l15 < 10) ? fcW[(k0+1)*10 + l15] : 0.f;
    acc = __builtin_amdgcn_wmma_f32_16x16x4_f32(false, a, false, b, (short)0, acc, false, false);
  }
  if (l15 < 10) {
#pragma unroll
    for (int r = 0; r < 8; ++r)
      out[(r + hi*8)*10 + l15] = acc[r];
  }
}

// ---------------- host ----------------

extern "C" void kernel_launch(void* const* d_in, const int* in_sizes, int n_in,
                              void* d_out, int out_size, void* d_ws, size_t ws_size,
                              hipStream_t stream) {
  (void)in_sizes; (void)n_in; (void)out_size; (void)ws_size;
  const float* x     = (const float*)d_in[0];
  const float* pos   = (const float*)d_in[1];
  const int*   batch = (const int*)d_in[2];
  const int*   ei    = (const int*)d_in[3];
  const float* ea    = (const float*)d_in[4];
  const int* srcI = ei;
  const int* dstI = ei + EE;
  const float *W[8], *gv[8], *bv[8];
  for (int i = 0; i < 8; ++i) {
    W[i]  = (const float*)d_in[5 + 3*i];
    gv[i] = (const float*)d_in[6 + 3*i];
    bv[i] = (const float*)d_in[7 + 3*i];
  }
  const float* fcW = (const float*)d_in[29];
  float* out = (float*)d_out;

  char* ws = (char*)d_ws;
  size_t off = 0;
  auto alloc = [&](size_t bytes) -> char* {
    char* p = ws + off;
    off += (bytes + 255) & ~(size_t)255;
    return p;
  };
  float*    bas   = (float*)alloc((size_t)EE*4*4);
  float*    bas2  = (float*)alloc((size_t)EE*4*4);
  float*    emask = (float*)alloc((size_t)EE*4);
  int*      csrc  = (int*)alloc((size_t)EE*4);
  int*      cdst  = (int*)alloc((size_t)EE*4);
  int*      cid   = (int*)alloc((size_t)NN*4);
  float*    deg   = (float*)alloc((size_t)NN*4);
  float*    hA    = (float*)alloc((size_t)NN*32*4);
  float*    hB    = (float*)alloc((size_t)NN*32*4);
  float*    Y     = (float*)alloc((size_t)NN*128*4);     // reused for pooled transforms
  float*    agg   = (float*)alloc((size_t)NN*32*4);      // reused for pooled aggregate
  unsigned* hash  = (unsigned*)alloc((size_t)HSIZE*4);
  float*    cnt   = (float*)alloc((size_t)SS*4);
  float*    ppx   = (float*)alloc((size_t)SS*4);
  float*    ppy   = (float*)alloc((size_t)SS*4);
  float*    wv    = (float*)alloc((size_t)SS*4);
  float*    deg2  = (float*)alloc((size_t)SS*4);
  unsigned* xpord = (unsigned*)alloc((size_t)SS*32*4);
  float*    xp    = (float*)alloc((size_t)SS*32*4);
  float*    hpA   = (float*)alloc((size_t)SS*32*4);
  float*    hpB   = (float*)alloc((size_t)SS*32*4);
  float*    stats = (float*)alloc(128*4);
  unsigned* mxb   = (unsigned*)alloc(256);
  float*    tw    = (float*)alloc(256);
  float*    gmn   = (float*)alloc(16*32*4);
  float*    gmd   = (float*)alloc(16*4);

  // per-call re-init of all accumulators (graph-capture safe)
  hipMemsetAsync(deg,   0,    (size_t)NN*4, stream);
  hipMemsetAsync(cnt,   0,    (size_t)SS*4, stream);
  hipMemsetAsync(ppx,   0,    (size_t)SS*4, stream);
  hipMemsetAsync(ppy,   0,    (size_t)SS*4, stream);
  hipMemsetAsync(deg2,  0,    (size_t)SS*4, stream);
  hipMemsetAsync(xpord, 0,    (size_t)SS*32*4, stream);
  hipMemsetAsync(hash,  0xFF, (size_t)HSIZE*4, stream);
  hipMemsetAsync(mxb,   0,    4, stream);
  hipMemsetAsync(tw,    0,    4, stream);
  hipMemsetAsync(gmn,   0,    16*32*4, stream);
  hipMemsetAsync(gmd,   0,    16*4, stream);

  const int TB = 256;
  k_basis_deg<<<(EE+TB-1)/TB, TB, 0, stream>>>(ea, dstI, bas, deg, EE);
  k_cells    <<<(NN+TB-1)/TB, TB, 0, stream>>>(pos, batch, cid, cnt, ppx, ppy, NN);
  k_pp       <<<(SS+TB-1)/TB, TB, 0, stream>>>(cnt, ppx, ppy, wv, tw, SS);
  k_edgemask <<<(EE+TB-1)/TB, TB, 0, stream>>>(srcI, dstI, cid, csrc, cdst, emask, hash, ppx, ppy, mxb, EE);
  k_bas2     <<<(EE+TB-1)/TB, TB, 0, stream>>>(csrc, cdst, emask, ppx, ppy, mxb, bas2, deg2, EE);

  auto transform = [&](const float* Xin, int rows, int cin, int cout, const float* Wl) {
    dim3 tg((rows/16 + 3)/4, (4*cout)/16);
    if      (cin==10 && cout== 8) k_transform_t<10, 8><<<tg,128,0,stream>>>(Xin, Wl, Y, rows);
    else if (cin== 8 && cout==16) k_transform_t< 8,16><<<tg,128,0,stream>>>(Xin, Wl, Y, rows);
    else if (cin==16 && cout==16) k_transform_t<16,16><<<tg,128,0,stream>>>(Xin, Wl, Y, rows);
    else if (cin==16 && cout==32) k_transform_t<16,32><<<tg,128,0,stream>>>(Xin, Wl, Y, rows);
    else                          k_transform_t<32,32><<<tg,128,0,stream>>>(Xin, Wl, Y, rows);
  };
  auto edgeagg = [&](const int* sE, const int* dE, const float* basE, const float* emk, int cout) {
    const float4* b4 = (const float4*)basE;
    if (emk)            k_edgeagg_t<32,true ><<<4096,256,0,stream>>>(sE,dE,b4,Y,agg,emk,EE);
    else if (cout== 8)  k_edgeagg_t< 8,false><<<4096,256,0,stream>>>(sE,dE,b4,Y,agg,nullptr,EE);
    else if (cout==16)  k_edgeagg_t<16,false><<<4096,256,0,stream>>>(sE,dE,b4,Y,agg,nullptr,EE);
    else                k_edgeagg_t<32,false><<<4096,256,0,stream>>>(sE,dE,b4,Y,agg,nullptr,EE);
  };
  auto layer = [&](const float* Xin, int rows, int cin, int cout,
                   const float* Wl, const float* gl, const float* bl,
                   const int* sE, const int* dE, const float* basE, const float* emk,
                   const float* degv, const float* wvv, const float* twp,
                   const float* sc, float* hout) {
    int lc = (cout==8) ? 3 : (cout==16 ? 4 : 5);
    transform(Xin, rows, cin, cout, Wl);
    hipMemsetAsync(agg, 0, (size_t)rows*cout*4, stream);
    edgeagg(sE, dE, basE, emk, cout);
    hipMemsetAsync(stats, 0, 128*4, stream);
    int nel = rows*cout;
    k_post1<<<(nel+TB-1)/TB, TB, 0, stream>>>(agg, degv, hout, stats, wvv, rows, cout, lc);
    k_post2<<<(nel+TB-1)/TB, TB, 0, stream>>>(hout, stats, gl, bl, sc, twp, (float)rows, rows, cout, lc);
  };

  // node-level layers
  layer(x,  NN, 10,  8, W[0], gv[0], bv[0], srcI, dstI, bas, nullptr, deg, nullptr, nullptr, nullptr, hA);
  layer(hA, NN,  8, 16, W[1], gv[1], bv[1], srcI, dstI, bas, nullptr, deg, nullptr, nullptr, nullptr, hB);
  layer(hB, NN, 16, 16, W[2], gv[2], bv[2], srcI, dstI, bas, nullptr, deg, nullptr, nullptr, nullptr, hA); // sc
  layer(hA, NN, 16, 16, W[3], gv[3], bv[3], srcI, dstI, bas, nullptr, deg, nullptr, nullptr, nullptr, hB);
  layer(hB, NN, 16, 16, W[4], gv[4], bv[4], srcI, dstI, bas, nullptr, deg, nullptr, nullptr, hA, hB);     // +sc
  layer(hB, NN, 16, 32, W[5], gv[5], bv[5], srcI, dstI, bas, nullptr, deg, nullptr, nullptr, nullptr, hA);

  // grid pooling (segment max)
  k_poolmax   <<<((NN*32)+TB-1)/TB, TB, 0, stream>>>(hA, cid, xpord, NN);
  k_pooldecode<<<((SS*32)+TB-1)/TB, TB, 0, stream>>>(xpord, cnt, xp, SS);

  // pooled-graph layers (weighted BN, masked edges)
  layer(xp,  SS, 32, 32, W[6], gv[6], bv[6], csrc, cdst, bas2, emask, deg2, wv, tw, nullptr, hpA);
  layer(hpA, SS, 32, 32, W[7], gv[7], bv[7], csrc, cdst, bas2, emask, deg2, wv, tw, nullptr, hpB);

  // global mean pool + FC
  k_gm<<<((SS*32)+TB-1)/TB, TB, 0, stream>>>(hpB, wv, gmn, gmd, SS);
  k_fc<<<1, 32, 0, stream>>>(gmn, gmd, fcW, out);
}